// PG_unfold_32246614459006
// MI455X (gfx1250) — compile-verified
//
#include <hip/hip_runtime.h>
#include <hip/hip_bf16.h>

typedef float v2f __attribute__((ext_vector_type(2)));
typedef float v8f __attribute__((ext_vector_type(8)));

#define BATCH 4096
#define MDIM  128
#define NDIM  96
#define LITER 20
#define HT_S  132   // H^T row stride (floats): 132 % 64 = 4 -> conflict-free b64 frag loads
#define G_S   97    // HTH row stride (floats): 97 % 64 = 33, odd -> conflict-free matvec

__global__ void pg_zero_loss_kernel(float* loss) { *loss = 0.f; }

__global__ __launch_bounds__(128) void pg_unfold_kernel(
    const float* __restrict__ x_ini, const float* __restrict__ y,
    const float* __restrict__ H,     const float* __restrict__ xt,
    const float* __restrict__ grad_ss, const float* __restrict__ extra_ss,
    const float* __restrict__ gamma1, float* __restrict__ out)
{
    __shared__ float sHT[NDIM * HT_S];   // H^T : [n][m]
    __shared__ float sG [NDIM * G_S];    // HTH : [n][n]
    __shared__ float sy [MDIM];
    __shared__ float sHTy[NDIM];
    __shared__ float sx [NDIM];
    __shared__ float syx[NDIM];
    __shared__ float sred[128];

    const int tid  = threadIdx.x;
    const int b    = blockIdx.x;
    const int lane = tid & 31;           // wave32
    const int wave = tid >> 5;

    // ---- load H coalesced (float4), scatter transposed into LDS ----
    const float4* Hg4 = (const float4*)(H + (size_t)b * (MDIM * NDIM));
    for (int i = tid; i < (MDIM * NDIM) / 4; i += 128) {
        float4 v = Hg4[i];
        int m  = i / (NDIM / 4);         // 24 float4 per row of H
        int n0 = (i % (NDIM / 4)) * 4;
        sHT[(n0 + 0) * HT_S + m] = v.x;
        sHT[(n0 + 1) * HT_S + m] = v.y;
        sHT[(n0 + 2) * HT_S + m] = v.z;
        sHT[(n0 + 3) * HT_S + m] = v.w;
    }
    if (tid < MDIM) sy[tid] = y[(size_t)b * MDIM + tid];
    if (tid < NDIM) {
        float xi = x_ini[(size_t)b * NDIM + tid];
        sx[tid] = xi; syx[tid] = xi;
    }
    __syncthreads();

    // ---- HTy (tiny, vector path; waves 0-2 fully active, wave 3 fully idle) ----
    if (tid < NDIM) {
        float acc = 0.f;
        #pragma unroll 4
        for (int m = 0; m < MDIM; ++m) acc += sHT[tid * HT_S + m] * sy[m];
        sHTy[tid] = acc;
    }

    // ---- HTH = H^T H via V_WMMA_F32_16X16X4_F32, upper-triangular tiles only ----
    {
        const int l16 = lane & 15;
        const int kh  = lane >> 4;       // selects K-pair {0,1} vs {2,3} of each 16x4 step
        for (int idx = wave; idx < 21; idx += 4) {       // wave-uniform tile loop
            int ti = 0, rem = idx;
            while (rem >= 6 - ti) { rem -= 6 - ti; ++ti; }
            int tj = ti + rem;
            const v2f* A2 = (const v2f*)&sHT[(ti * 16 + l16) * HT_S];
            const v2f* B2 = (const v2f*)&sHT[(tj * 16 + l16) * HT_S];
            v8f c = {};
            #pragma unroll
            for (int m2 = 0; m2 < MDIM / 2; m2 += 2) {   // 32 chained WMMAs over M=128
                v2f a  = A2[m2 + kh];
                v2f bb = B2[m2 + kh];
                c = __builtin_amdgcn_wmma_f32_16x16x4_f32(
                        false, a, false, bb, (short)0, c, false, false);
            }
            // C/D layout: VGPR v holds rows v (lanes 0-15) and v+8 (lanes 16-31)
            int rb = kh * 8;
            #pragma unroll
            for (int v = 0; v < 8; ++v) {
                sG[(ti * 16 + rb + v) * G_S + (tj * 16 + l16)] = c[v];
                if (ti != tj)                              // mirror (HTH symmetric)
                    sG[(tj * 16 + l16) * G_S + (ti * 16 + rb + v)] = c[v];
            }
        }
    }
    __syncthreads();

    // ---- 20 FISTA iterations, fully in LDS ----
    for (int it = 0; it < LITER; ++it) {
        float ply = 0.f, ynew = 0.f;
        if (tid < NDIM) {
            float g = 0.f;
            #pragma unroll 4
            for (int j = 0; j < NDIM; ++j) g += sG[tid * G_S + j] * syx[j];
            g = 2.f * (g - sHTy[tid]);
            float gs = grad_ss[it], es = extra_ss[it], gm = gamma1[it];
            float xb = syx[tid] - gs * g;
            // argmin over {-2,0,2}, first-index tie-break like jnp.argmin
            float center = (xb <= -1.f) ? -2.f : ((xb <= 1.f) ? 0.f : 2.f);
            ply  = 2.f / (1.f + expf(-gm * (xb - center))) - 1.f;
            ynew = ply + es * (ply - sx[tid]);
        }
        __syncthreads();
        if (tid < NDIM) { sx[tid] = ply; syx[tid] = ynew; }
        __syncthreads();
    }

    // ---- write x, accumulate loss ----
    float part = 0.f;
    if (tid < NDIM) {
        float xv = sx[tid];
        out[(size_t)b * NDIM + tid] = xv;
        float d = xv - xt[(size_t)b * NDIM + tid];
        part = d * d;
    }
    sred[tid] = part;
    __syncthreads();
    for (int s = 64; s > 0; s >>= 1) {
        if (tid < s) sred[tid] += sred[tid + s];
        __syncthreads();
    }
    if (tid == 0) unsafeAtomicAdd(out + (size_t)BATCH * NDIM, sred[0]);
}

extern "C" void kernel_launch(void* const* d_in, const int* in_sizes, int n_in,
                              void* d_out, int out_size, void* d_ws, size_t ws_size,
                              hipStream_t stream) {
    const float* x_ini    = (const float*)d_in[0];
    const float* y        = (const float*)d_in[1];
    const float* H        = (const float*)d_in[2];
    const float* xt       = (const float*)d_in[3];
    const float* grad_ss  = (const float*)d_in[4];
    const float* extra_ss = (const float*)d_in[5];
    const float* gamma1   = (const float*)d_in[6];
    float* out = (float*)d_out;

    // loss slot must be zeroed every call (harness poisons once, graph replays after)
    pg_zero_loss_kernel<<<1, 1, 0, stream>>>(out + (size_t)BATCH * NDIM);
    pg_unfold_kernel<<<BATCH, 128, 0, stream>>>(
        x_ini, y, H, xt, grad_ss, extra_ss, gamma1, out);
}